// ProvenGSA_31817117729247
// MI455X (gfx1250) — compile-verified
//
#include <hip/hip_runtime.h>
#include <hip/hip_bf16.h>

// ---------------------------------------------------------------------------
// Gaussian Sparse Attention forward for MI455X (gfx1250, wave32, WMMA).
//
// All GEMM stages use V_WMMA_F32_16X16X4_F32 (fp32 in/out). Problem is
// ~14 GFLOP vs ~550 MB mandatory HBM traffic (attn [B,T,T,H] is a returned
// output) -> near the 23.3 TB/s memory roof, so fp32 fidelity is free.
// Register blocking: each wave computes a 16x64 output strip (4 accumulators)
// so one A fragment feeds 4 WMMAs. K loop is chunked by 64 floats with one
// unconditional speculative global_prefetch_b8 per chunk (no branch: specu-
// lative prefetch translation failures are silently dropped per the ISA).
// ---------------------------------------------------------------------------

#define DEV static __device__ __forceinline__

constexpr int B  = 4;
constexpr int T  = 1024;
constexpr int D  = 512;
constexpr int H  = 8;
constexpr int S  = 16;
constexpr int HD = 64;       // D / H
constexpr float PRUNE = 0.02f;

typedef __attribute__((ext_vector_type(2))) float v2f;
typedef __attribute__((ext_vector_type(8))) float v8f;

// ---- WMMA f32 16x16x4: D = A(16x4) * B(4x16) + C --------------------------
DEV v8f wmma4(v2f a, v2f b, v8f c) {
  return __builtin_amdgcn_wmma_f32_16x16x4_f32(
      false, a, false, b, (short)0, c, false, false);
}

// A-fragment, or transposed-B fragment from [rows x K] row-major storage.
// ISA 7.12.2 (32-bit A 16x4): lanes 0-15 hold K=k,k+1; lanes 16-31 hold K=k+2,k+3.
DEV v2f load_frag_nk(const float* __restrict__ base, int ld, int r0, int k, int lane) {
  const float* p = base + (size_t)(r0 + (lane & 15)) * ld + (k + ((lane >> 4) << 1));
  v2f f; f.x = p[0]; f.y = p[1];
  return f;
}

// B-fragment from natural [K x N] row-major storage (e.g. the V matrix).
// VGPR0 = rows k (lanes 0-15) / k+2 (lanes 16-31); VGPR1 = rows k+1 / k+3.
DEV v2f load_frag_kn(const float* __restrict__ base, int ld, int k, int n0, int lane) {
  const float* p = base + (size_t)(k + ((lane >> 4) << 1)) * ld + n0 + (lane & 15);
  v2f f; f.x = p[0]; f.y = p[ld];
  return f;
}

// C/D tile store: VGPR r -> rows M=r (lanes 0-15) and M=8+r (lanes 16-31).
DEV void store_tile(float* __restrict__ base, int ld, int r0, int n0, int lane, v8f c) {
  int col   = n0 + (lane & 15);
  int rbase = r0 + ((lane >> 4) << 3);
#pragma unroll
  for (int r = 0; r < 8; ++r) base[(size_t)(rbase + r) * ld + col] = c[r];
}

// ---------------------------------------------------------------------------
// Kernel 1: tiny parameter prep (centers, inv_var, pruned amps, |c|^2, alive)
// ---------------------------------------------------------------------------
__global__ void k_prep(const float* __restrict__ base_centers,
                       const float* __restrict__ deltas,
                       const float* __restrict__ log_scales,
                       const float* __restrict__ log_amps,
                       const float* __restrict__ msp,
                       float* __restrict__ centers, float* __restrict__ inv_var,
                       float* __restrict__ eamps,   float* __restrict__ c2,
                       float* __restrict__ alive) {
  int tid = threadIdx.x;                       // 128 threads == H*S
  float move = 0.2f / (1.0f + __expf(-msp[0]));
  if (tid < H * S) {
    float acc = 0.0f;
    for (int d = 0; d < HD; ++d) {
      float c = base_centers[tid * HD + d] + deltas[tid * HD + d] * move;
      centers[tid * HD + d] = c;
      acc += c * c;
    }
    c2[tid] = acc;
    float sc = fminf(fmaxf(__expf(log_scales[tid]), 1e-4f), 3.0f);
    inv_var[tid] = 1.0f / (sc * sc + 1e-8f);
    float a = fminf(fmaxf(__expf(log_amps[tid]), 1e-6f), 10.0f);
    eamps[tid] = (a > PRUNE) ? a : 0.0f;
  }
  __syncthreads();
  if (tid < H) {
    float s = 0.0f;
    for (int j = 0; j < S; ++j) s += eamps[tid * S + j];
    alive[tid] = (s >= 1e-8f) ? 1.0f : 0.0f;
  }
}

// ---------------------------------------------------------------------------
// Kernel 2: QKV projection  C[4096,1536] = x[4096,512] @ Wqkv[1536,512]^T
// One wave per 16x64 strip (4 accumulators). Scatter into q/k/v [B,H,T,HD].
// ---------------------------------------------------------------------------
__global__ void k_qkv(const float* __restrict__ x, const float* __restrict__ Wqkv,
                      float* __restrict__ q, float* __restrict__ kmat,
                      float* __restrict__ v) {
  int lane = threadIdx.x & 31;
  int m0 = blockIdx.x * 16;          // token rows (b*T + t)
  int n0 = blockIdx.y * 64;          // output feature cols, 64-wide strip
  const float* arow = x + (size_t)(m0 + (lane & 15)) * D;
  v8f acc[4] = {};
  for (int kc = 0; kc < D; kc += 64) {
    __builtin_prefetch(arow + kc + 64, 0, 1);   // speculative, branch-free
#pragma unroll
    for (int k4 = 0; k4 < 64; k4 += 4) {
      int kk = kc + k4;
      v2f a = load_frag_nk(x, D, m0, kk, lane);
#pragma unroll
      for (int nt = 0; nt < 4; ++nt) {
        v2f b = load_frag_nk(Wqkv, D, n0 + nt * 16, kk, lane);
        acc[nt] = wmma4(a, b, acc[nt]);
      }
    }
  }
  // 64-wide strip stays inside one (which, h) block: n0 is 64-aligned.
  int which = n0 >> 9;               // 0=q,1=k,2=v
  int h     = (n0 >> 6) & (H - 1);
  float* dst = (which == 0) ? q : (which == 1) ? kmat : v;
  int rbase = m0 + ((lane >> 4) << 3);
#pragma unroll
  for (int nt = 0; nt < 4; ++nt) {
    int hd = nt * 16 + (lane & 15);
#pragma unroll
    for (int r = 0; r < 8; ++r) {
      int m  = rbase + r;
      int bb = m >> 10;              // / T
      int t  = m & (T - 1);
      dst[((size_t)(bb * H + h) * T + t) * HD + hd] = acc[nt][r];
    }
  }
}

// ---------------------------------------------------------------------------
// Kernel 3: per-row squared norms of q and k ([B,H,T])
// ---------------------------------------------------------------------------
__global__ void k_norms(const float* __restrict__ q, const float* __restrict__ kmat,
                        float* __restrict__ q2, float* __restrict__ k2) {
  int i = blockIdx.x * blockDim.x + threadIdx.x;
  if (i >= B * H * T) return;
  const float* pq = q    + (size_t)i * HD;
  const float* pk = kmat + (size_t)i * HD;
  float sq = 0.0f, sk = 0.0f;
  for (int d = 0; d < HD; ++d) {
    float a = pq[d]; sq += a * a;
    float b = pk[d]; sk += b * b;
  }
  q2[i] = sq; k2[i] = sk;
}

// ---------------------------------------------------------------------------
// Kernel 4: Gaussian weights. WMMA computes q.c (K=HD=64, N=S=16), then
// w = exp(-0.5 * min(|q|^2 - 2 q.c + |c|^2, 50) * inv_var) [* amps for K-side]
// ---------------------------------------------------------------------------
__global__ void k_gauss(const float* __restrict__ src, const float* __restrict__ n2,
                        const float* __restrict__ centers,
                        const float* __restrict__ inv_var,
                        const float* __restrict__ eamps,
                        const float* __restrict__ c2,
                        float* __restrict__ dst, int applyAmps) {
  int lane = threadIdx.x & 31;
  int m0 = blockIdx.x * 16;          // t tile
  int bh = blockIdx.y;               // b*H + h
  int h  = bh & (H - 1);
  const float* A = src + (size_t)bh * T * HD;       // [T,HD]
  const float* C = centers + h * S * HD;            // [S,HD] (transposed-B role)
  v8f acc = {};
#pragma unroll
  for (int kk = 0; kk < HD; kk += 4) {
    v2f a = load_frag_nk(A, HD, m0, kk, lane);
    v2f b = load_frag_nk(C, HD, 0,  kk, lane);
    acc = wmma4(a, b, acc);
  }
  int s    = lane & 15;
  float iv = inv_var[h * S + s];
  float cc = c2[h * S + s];
  float am = applyAmps ? eamps[h * S + s] : 1.0f;
  int rbase = m0 + ((lane >> 4) << 3);
#pragma unroll
  for (int r = 0; r < 8; ++r) {
    int t = rbase + r;
    float d = n2[(size_t)bh * T + t] - 2.0f * acc[r] + cc;
    d = fminf(d, 50.0f);
    dst[((size_t)bh * T + t) * S + s] = am * __expf(-0.5f * d * iv);
  }
}

// ---------------------------------------------------------------------------
// Kernel 5: scores[b,h,i,j] = sum_s qw[i,s] * kwa[j,s]   (K = S = 16)
// One wave per 16x64 strip of j (4 accumulators).
// ---------------------------------------------------------------------------
__global__ void k_scores(const float* __restrict__ qw, const float* __restrict__ kwa,
                         float* __restrict__ scores) {
  int lane = threadIdx.x & 31;
  int i0 = blockIdx.x * 16;
  int j0 = blockIdx.y * 64;
  int bh = blockIdx.z;
  const float* A  = qw  + (size_t)bh * T * S;
  const float* Bm = kwa + (size_t)bh * T * S;
  v8f acc[4] = {};
#pragma unroll
  for (int kk = 0; kk < S; kk += 4) {
    v2f a = load_frag_nk(A, S, i0, kk, lane);
#pragma unroll
    for (int nt = 0; nt < 4; ++nt) {
      v2f b = load_frag_nk(Bm, S, j0 + nt * 16, kk, lane);
      acc[nt] = wmma4(a, b, acc[nt]);
    }
  }
  float* Cb = scores + (size_t)bh * T * T;
#pragma unroll
  for (int nt = 0; nt < 4; ++nt) store_tile(Cb, T, i0, j0 + nt * 16, lane, acc[nt]);
}

// ---------------------------------------------------------------------------
// Kernel 6: softmax over j; alive->eye fallback; writes normalized attn back
// in-place ([B,H,T,T], coalesced, feeds attn@V) and to d_out in bijh layout.
// ---------------------------------------------------------------------------
__global__ void k_softmax(float* __restrict__ scores, const float* __restrict__ alive,
                          const float* __restrict__ temp_p,
                          float* __restrict__ attn_out) {
  __shared__ float red[256];
  int rid = blockIdx.x;              // [0, B*H*T)
  int tid = threadIdx.x;
  int bh = rid >> 10;                // / T
  int i  = rid & (T - 1);
  int b  = bh >> 3, h = bh & (H - 1);
  float invt = 1.0f / fminf(fmaxf(temp_p[0], 0.1f), 10.0f);
  bool al = alive[h] > 0.5f;
  float* row = scores + (size_t)rid * T;

  float vals[4];
  float mx = -1e30f;
#pragma unroll
  for (int it = 0; it < 4; ++it) {
    int j = tid + it * 256;
    float sv = al ? row[j] : ((i == j) ? 1.0f : 0.0f);
    float lg = sv * invt + 1e-8f;
    vals[it] = lg;
    mx = fmaxf(mx, lg);
  }
  red[tid] = mx; __syncthreads();
  for (int off = 128; off > 0; off >>= 1) {
    if (tid < off) red[tid] = fmaxf(red[tid], red[tid + off]);
    __syncthreads();
  }
  mx = red[0]; __syncthreads();

  float sum = 0.0f;
#pragma unroll
  for (int it = 0; it < 4; ++it) { vals[it] = __expf(vals[it] - mx); sum += vals[it]; }
  red[tid] = sum; __syncthreads();
  for (int off = 128; off > 0; off >>= 1) {
    if (tid < off) red[tid] += red[tid + off];
    __syncthreads();
  }
  float inv = 1.0f / red[0];

  float* ob = attn_out + ((size_t)b * T + i) * T * H + h;   // [b,i,j,h]
#pragma unroll
  for (int it = 0; it < 4; ++it) {
    int j = tid + it * 256;
    float a = vals[it] * inv;
    row[j] = a;
    ob[(size_t)j * H] = a;
  }
}

// ---------------------------------------------------------------------------
// Kernel 7: o[b,t,h*HD+hd] = sum_j attn[b,h,t,j] * v[b,h,j,hd]   (K = T)
// One wave per 16-row tile x full HD=64 strip (4 accumulators).
// ---------------------------------------------------------------------------
__global__ void k_av(const float* __restrict__ attn, const float* __restrict__ v,
                     float* __restrict__ o) {
  int lane = threadIdx.x & 31;
  int i0 = blockIdx.x * 16;
  int bh = blockIdx.y;
  int b  = bh >> 3, h = bh & (H - 1);
  const float* A  = attn + (size_t)bh * T * T;      // [T,T]
  const float* Bv = v    + (size_t)bh * T * HD;     // [T,HD] natural K x N
  const float* arow = A + (size_t)(i0 + (lane & 15)) * T;
  v8f acc[4] = {};
  for (int kc = 0; kc < T; kc += 64) {
    __builtin_prefetch(arow + kc + 64, 0, 1);   // speculative, branch-free
#pragma unroll
    for (int k4 = 0; k4 < 64; k4 += 4) {
      int kk = kc + k4;
      v2f a = load_frag_nk(A, T, i0, kk, lane);
#pragma unroll
      for (int nt = 0; nt < 4; ++nt) {
        v2f bf = load_frag_kn(Bv, HD, kk, nt * 16, lane);
        acc[nt] = wmma4(a, bf, acc[nt]);
      }
    }
  }
  int rbase = i0 + ((lane >> 4) << 3);
#pragma unroll
  for (int nt = 0; nt < 4; ++nt) {
    int col = h * HD + nt * 16 + (lane & 15);
#pragma unroll
    for (int r = 0; r < 8; ++r) {
      int t = rbase + r;
      o[((size_t)(b * T + t)) * D + col] = acc[nt][r];
    }
  }
}

// ---------------------------------------------------------------------------
// Kernel 8: out[4096,512] = o[4096,512] @ Wout[512,512]^T  (16x64 strips)
// ---------------------------------------------------------------------------
__global__ void k_out(const float* __restrict__ o, const float* __restrict__ Wout,
                      float* __restrict__ out) {
  int lane = threadIdx.x & 31;
  int m0 = blockIdx.x * 16, n0 = blockIdx.y * 64;
  const float* arow = o + (size_t)(m0 + (lane & 15)) * D;
  v8f acc[4] = {};
  for (int kc = 0; kc < D; kc += 64) {
    __builtin_prefetch(arow + kc + 64, 0, 1);   // speculative, branch-free
#pragma unroll
    for (int k4 = 0; k4 < 64; k4 += 4) {
      int kk = kc + k4;
      v2f a = load_frag_nk(o, D, m0, kk, lane);
#pragma unroll
      for (int nt = 0; nt < 4; ++nt) {
        v2f b = load_frag_nk(Wout, D, n0 + nt * 16, kk, lane);
        acc[nt] = wmma4(a, b, acc[nt]);
      }
    }
  }
#pragma unroll
  for (int nt = 0; nt < 4; ++nt) store_tile(out, D, m0, n0 + nt * 16, lane, acc[nt]);
}

// ---------------------------------------------------------------------------
extern "C" void kernel_launch(void* const* d_in, const int* in_sizes, int n_in,
                              void* d_out, int out_size, void* d_ws, size_t ws_size,
                              hipStream_t stream) {
  const float* x            = (const float*)d_in[0];
  const float* Wqkv         = (const float*)d_in[1];
  const float* Wout         = (const float*)d_in[2];
  const float* base_centers = (const float*)d_in[3];
  const float* deltas       = (const float*)d_in[4];
  const float* log_scales   = (const float*)d_in[5];
  const float* log_amps     = (const float*)d_in[6];
  const float* msp          = (const float*)d_in[7];
  const float* temp         = (const float*)d_in[8];

  float* ws = (float*)d_ws;
  size_t off = 0;
  float* centers = ws + off; off += (size_t)H * S * HD;
  float* inv_var = ws + off; off += H * S;
  float* eamps   = ws + off; off += H * S;
  float* c2      = ws + off; off += H * S;
  float* alive   = ws + off; off += 32;
  float* q       = ws + off; off += (size_t)B * H * T * HD;
  float* kmat    = ws + off; off += (size_t)B * H * T * HD;
  float* v       = ws + off; off += (size_t)B * H * T * HD;
  float* q2      = ws + off; off += (size_t)B * H * T;
  float* k2      = ws + off; off += (size_t)B * H * T;
  float* qw      = ws + off; off += (size_t)B * H * T * S;
  float* kwa     = ws + off; off += (size_t)B * H * T * S;
  float* o       = ws + off; off += (size_t)B * T * D;
  float* scores  = ws + off; off += (size_t)B * H * T * T;   // 134 MB

  float* out_proj = (float*)d_out;                       // [B,T,D]
  float* attn_out = out_proj + (size_t)B * T * D;        // [B,T,T,H]

  // 1. parameter prep
  k_prep<<<dim3(1), dim3(128), 0, stream>>>(base_centers, deltas, log_scales,
                                            log_amps, msp, centers, inv_var,
                                            eamps, c2, alive);
  // 2. QKV projection (WMMA, 16x64 strips)
  k_qkv<<<dim3((B * T) / 16, (3 * D) / 64), dim3(32), 0, stream>>>(x, Wqkv, q, kmat, v);
  // 3. row norms
  k_norms<<<dim3((B * H * T + 255) / 256), dim3(256), 0, stream>>>(q, kmat, q2, k2);
  // 4. Gaussian weights (WMMA for q.c / k.c)
  k_gauss<<<dim3(T / 16, B * H), dim3(32), 0, stream>>>(q, q2, centers, inv_var,
                                                        eamps, c2, qw, 0);
  k_gauss<<<dim3(T / 16, B * H), dim3(32), 0, stream>>>(kmat, k2, centers, inv_var,
                                                        eamps, c2, kwa, 1);
  // 5. scores (WMMA, K=16, 16x64 strips)
  k_scores<<<dim3(T / 16, T / 64, B * H), dim3(32), 0, stream>>>(qw, kwa, scores);
  // 6. softmax + attn output materialization
  k_softmax<<<dim3(B * H * T), dim3(256), 0, stream>>>(scores, alive, temp, attn_out);
  // 7. attn @ V (WMMA, K=1024, 16x64 strips)
  k_av<<<dim3(T / 16, B * H), dim3(32), 0, stream>>>(scores, v, o);
  // 8. output projection (WMMA, 16x64 strips)
  k_out<<<dim3((B * T) / 16, D / 64), dim3(32), 0, stream>>>(o, Wout, out_proj);
}